// BoxEmbeddingCBM_14877766713983
// MI455X (gfx1250) — compile-verified
//
#include <hip/hip_runtime.h>

// ---------------------------------------------------------------------------
// BoxEmbeddingCBM on MI455X (gfx1250, wave32).
//  - theta = features(4096x2048) @ Wp^T(2048x1024): f32 WMMA 16x16x4 (exact
//    f32 accumulate; 17.2 GFLOP), fed by Tensor-Data-Mover DMA into LDS with
//    double buffering (TENSORcnt-tracked, per ISA cdna5 ch.7/8).
//  - Pairwise box intersections (B*K*K*D = 134M lanes of logaddexp/softplus):
//    per-row LDS-staged VALU kernel.
//  - cond output 67MB -> ~3us at 23.3 TB/s; all weights resident in 192MB L2.
// ---------------------------------------------------------------------------

typedef __attribute__((ext_vector_type(2))) float v2f;
typedef __attribute__((ext_vector_type(8))) float v8f;
typedef __attribute__((ext_vector_type(4))) unsigned int u32x4;
typedef __attribute__((ext_vector_type(8))) int i32x8;
typedef __attribute__((ext_vector_type(4))) int i32x4;

#define Bsz    4096
#define Fdim   2048
#define Kc     64
#define Dd     8
#define N2D    16          // 2*D
#define Ncols  (Kc * N2D)  // 1024
#define KCHUNK 32
#define NCHUNK (Fdim / KCHUNK)   // 64
#define APITCH 33          // 32 data dwords + 1 pad dword per row (bank-conflict free)

__device__ __forceinline__ float softplus_f(float x) {
    return fmaxf(x, 0.f) + log1pf(__expf(-fabsf(x)));
}
__device__ __forceinline__ float sigmoid_f(float x) {
    return 1.f / (1.f + __expf(-x));
}

// ---------------------------------------------------------------------------
// TDM 2-D tile load: global -> LDS.  D# packed per CDNA5 ISA 8.3/8.4:
//  group0: [1:0]=count=1 | [63:32]=lds_addr | [120:64]=global_addr | [127:126]=type=2
//  group1: [17:16]=data_size(2=4B) | [20]=pad_en | [24:22]=pad_interval |
//          [31:25]=pad_amount | [79:48]=tensor_dim0 | [111:80]=tensor_dim1 |
//          [127:112]=tile_dim0 | [143:128]=tile_dim1 | [207:160]=dim0_stride
//  groups 2/3 (and trailing group): zero (<=2D tensor).
// 6-arg builtin form per this toolchain (clang-23 / therock-10.0 headers).
// ---------------------------------------------------------------------------
__device__ __forceinline__ void tdm_load_2d(unsigned lds_addr, const void* gptr,
                                            unsigned tile_d0, unsigned tile_d1,
                                            unsigned tensor_d0, unsigned tensor_d1,
                                            unsigned long long stride0,
                                            unsigned flags_w0)
{
    unsigned long long ga = (unsigned long long)(size_t)gptr;
    u32x4 g0;
    g0.x = 1u;                                   // count=1, user descriptor
    g0.y = lds_addr;                             // LDS byte address
    g0.z = (unsigned)ga;                         // global_addr[31:0]
    g0.w = (unsigned)((ga >> 32) & 0x1FFFFFFu)   // global_addr[56:32]
         | 0x80000000u;                          // type=2 ("image")
    i32x8 g1;
    g1[0] = (int)flags_w0;                                        // mask/ds/pad
    g1[1] = (int)((tensor_d0 & 0xFFFFu) << 16);                   // td0[15:0]
    g1[2] = (int)((tensor_d0 >> 16) | ((tensor_d1 & 0xFFFFu) << 16));
    g1[3] = (int)((tensor_d1 >> 16) | (tile_d0 << 16));
    g1[4] = (int)(tile_d1 & 0xFFFFu);                             // tile_dim1
    g1[5] = (int)(unsigned)(stride0 & 0xFFFFFFFFull);             // stride0 lo
    g1[6] = (int)(unsigned)((stride0 >> 32) & 0xFFFFull);         // stride0 hi
    g1[7] = 0;
    i32x4 z4 = {0, 0, 0, 0};
    i32x8 z8 = {0, 0, 0, 0, 0, 0, 0, 0};
    __builtin_amdgcn_tensor_load_to_lds(g0, g1, z4, z4, z8, 0);
}

// flags: data_size=4B
#define TDM_FLAGS_PLAIN 0x00020000u
// + pad_enable, pad_interval=4 (every 32 DWORDs), pad_amount=0 (1 DWORD)
#define TDM_FLAGS_PAD32 0x01120000u

// ---------------------------------------------------------------------------
// Kernel 1: theta[b, k*16+n] = sum_f features[b,f] * Wp[k,f,n] + bp[k,n]
// Block = 128 threads (4 waves) -> 64(M) x 16(N=one concept) tile.
// K-loop in chunks of 32, double-buffered TDM loads into LDS:
//   A tile 64x32 f32 (row pitch 33 dwords via TDM padding), B tile 32x16 f32.
// Wave w computes rows [w*16, w*16+16) with f32 WMMA 16x16x4 (8 per chunk).
//   A frag: lanes 0-15 -> M=lane, K={0,1}; lanes 16-31 -> K={2,3}
//   B frag: lanes 0-15 -> N=lane, K={0,1}; lanes 16-31 -> K={2,3}
//   C/D  : VGPR r, lanes 0-15 -> M=r; lanes 16-31 -> M=8+r
// ---------------------------------------------------------------------------
__global__ void __launch_bounds__(128) gemm_theta_wmma(
    const float* __restrict__ features,  // B x F
    const float* __restrict__ Wp,        // K x F x 16
    const float* __restrict__ bp,        // K x 16
    float* __restrict__ theta)           // B x 1024
{
    __shared__ float shA[2][64 * APITCH];   // 2 x 8448 B
    __shared__ float shB[2][KCHUNK * N2D];  // 2 x 2048 B

    const int tid    = threadIdx.x;
    const int lane   = tid & 31;
    const int waveId = tid >> 5;          // 0..3 -> M sub-tile
    const int kcpt   = blockIdx.x;        // concept index == N-tile
    const int m0     = blockIdx.y * 64;   // block M origin
    const int col    = lane & 15;
    const int khalf  = (lane >> 4) << 1;  // 0 or 2

    const float* __restrict__ Asrc = features + (size_t)m0 * Fdim;
    const float* __restrict__ Bsrc = Wp + (size_t)kcpt * Fdim * N2D;

    // prologue: chunk 0 -> buffer 0 (issued once by wave 0; TDM ignores EXEC)
    if (tid < 32) {
        tdm_load_2d((unsigned)(size_t)&shA[0][0], Asrc,
                    /*tile*/ KCHUNK, 64, /*tensor*/ Fdim, Bsz,
                    /*stride0*/ Fdim, TDM_FLAGS_PAD32);
        tdm_load_2d((unsigned)(size_t)&shB[0][0], Bsrc,
                    /*tile*/ KCHUNK * N2D, 1, /*tensor*/ Fdim * N2D, 1,
                    /*stride0*/ KCHUNK * N2D, TDM_FLAGS_PLAIN);
    }

    v8f acc = {};
    for (int c = 0; c < NCHUNK; ++c) {
        if (tid < 32) {
            if (c + 1 < NCHUNK) {   // prefetch next chunk into other buffer
                const int kk = (c + 1) * KCHUNK;
                const int nb = (c + 1) & 1;
                tdm_load_2d((unsigned)(size_t)&shA[nb][0], Asrc + kk,
                            KCHUNK, 64, Fdim, Bsz, Fdim, TDM_FLAGS_PAD32);
                tdm_load_2d((unsigned)(size_t)&shB[nb][0],
                            Bsrc + (size_t)kk * N2D,
                            KCHUNK * N2D, 1, Fdim * N2D, 1,
                            KCHUNK * N2D, TDM_FLAGS_PLAIN);
                // in-order per wave: <=2 outstanding means chunk c complete
                __builtin_amdgcn_s_wait_tensorcnt(2);
            } else {
                __builtin_amdgcn_s_wait_tensorcnt(0);
            }
        }
        __syncthreads();   // chunk c resident in LDS for all waves

        const float* __restrict__ Ab = &shA[c & 1][(waveId * 16 + col) * APITCH];
        const float* __restrict__ Bb = &shB[c & 1][0];
        #pragma unroll
        for (int kkk = 0; kkk < KCHUNK; kkk += 4) {
            v2f a, b2;
            a.x  = Ab[kkk + khalf];
            a.y  = Ab[kkk + khalf + 1];
            b2.x = Bb[(kkk + khalf)     * N2D + col];
            b2.y = Bb[(kkk + khalf + 1) * N2D + col];
            acc = __builtin_amdgcn_wmma_f32_16x16x4_f32(
                false, a, false, b2, (short)0, acc, false, false);
        }
        __syncthreads();   // everyone done with buf[c&1] before it is re-filled
    }

    const float bias  = bp[kcpt * N2D + col];
    const int   mbase = m0 + waveId * 16 + ((lane >> 4) << 3);  // +0 or +8
    #pragma unroll
    for (int r = 0; r < 8; ++r) {
        theta[(size_t)(mbase + r) * Ncols + kcpt * N2D + col] = acc[r] + bias;
    }
}

// ---------------------------------------------------------------------------
// Kernel 2: per (b,k): z/Z, concept sigmoid, lv_box, Wbox partial dot.
// Rewrites theta row in place as [z(0..7) | Z(0..7)].
// ---------------------------------------------------------------------------
__global__ void __launch_bounds__(64) concept_kernel(
    float* __restrict__ theta,           // in: theta; out: z|Z  (B x K x 16)
    const float* __restrict__ Wprob,     // K x 16
    const float* __restrict__ bprob,     // K
    const float* __restrict__ Wbox,      // 1024
    float* __restrict__ concept_out,     // B x K
    float* __restrict__ lvbox_ws,        // B x K
    float* __restrict__ boxdot_ws)       // B
{
    __shared__ float red[64];
    const int b = blockIdx.x;
    const int k = threadIdx.x;
    float* th = theta + ((size_t)b * Kc + k) * N2D;

    float z[Dd], Zt[Dd];
    float logit = bprob[k];
    float lv = 0.f;
    #pragma unroll
    for (int d = 0; d < Dd; ++d) {
        float zz = th[d];
        float Zz = zz + softplus_f(th[Dd + d]);
        z[d] = zz;  Zt[d] = Zz;
        logit += zz * Wprob[k * N2D + d] + Zz * Wprob[k * N2D + Dd + d];
        // beta = 1/VOL_TEMP = 2:  side = softplus(2*(Z-z))/2
        float y    = 2.f * (Zz - zz);
        float side = 0.5f * (fmaxf(y, 0.f) + log1pf(__expf(-fabsf(y))));
        lv += __logf(fmaxf(side, 1e-23f));
    }
    float p = sigmoid_f(logit);
    concept_out[(size_t)b * Kc + k] = p;
    lvbox_ws[(size_t)b * Kc + k]    = lv;

    float bdot = 0.f;
    #pragma unroll
    for (int d = 0; d < Dd; ++d) {
        bdot += z[d] * Wbox[k * N2D + d] + Zt[d] * Wbox[k * N2D + Dd + d];
        th[d]      = z[d];
        th[Dd + d] = Zt[d];    // store Z in place of raw theta
    }
    red[k] = bdot * p;
    __syncthreads();
    #pragma unroll
    for (int s = 32; s > 0; s >>= 1) {
        if (k < s) red[k] += red[k + s];
        __syncthreads();
    }
    if (k == 0) boxdot_ws[b] = red[0];
}

// ---------------------------------------------------------------------------
// Kernel 3: per b: all K*K=4096 pair intersections; cond output + Wrel dot.
// z|Z for the row staged in LDS (4KB) and reused across all pairs.
// ---------------------------------------------------------------------------
__global__ void __launch_bounds__(256) pair_kernel(
    const float* __restrict__ zZ,        // B x K x 16  (z|Z)
    const float* __restrict__ lvbox_ws,  // B x K
    const float* __restrict__ Wrel,      // K*K
    float* __restrict__ cond_out,        // B x K x K
    float* __restrict__ relsum_ws)       // B
{
    __shared__ float sh[Kc * N2D];   // 1024 floats of z|Z
    __shared__ float lvb[Kc];
    __shared__ float red[256];
    const int b = blockIdx.x;
    const int t = threadIdx.x;

    {   // cooperative load: 256 threads x float4 = 1024 floats
        const float4* src = (const float4*)(zZ + (size_t)b * Kc * N2D);
        ((float4*)sh)[t] = src[t];
        if (t < Kc) lvb[t] = lvbox_ws[(size_t)b * Kc + t];
    }
    __syncthreads();

    const float tt   = 0.1f;   // INT_TEMP
    const float invt = 10.f;
    float rel = 0.f;
    #pragma unroll
    for (int pi = 0; pi < 16; ++pi) {
        const int idx = t * 16 + pi;        // i = idx>>6 constant per thread
        const int i = idx >> 6, j = idx & 63;
        const float* bi = sh + i * N2D;
        const float* bj = sh + j * N2D;
        float lv = 0.f;
        #pragma unroll
        for (int d = 0; d < Dd; ++d) {
            float a = bi[d],      c = bj[d];       // z_i, z_j
            float A = bi[Dd + d], C = bj[Dd + d];  // Z_i, Z_j
            // z_int = t*logaddexp(zi/t, zj/t) = max + t*log1p(exp(-|dz|/t))
            float mx   = fmaxf(a, c);
            float zint = mx + tt * log1pf(__expf(-fabsf(a - c) * invt));
            zint = fmaxf(zint, mx);
            // Z_int = -t*logaddexp(-Zi/t,-Zj/t) = min - t*log1p(exp(-|dZ|/t))
            float mn   = fminf(A, C);
            float Zint = mn - tt * log1pf(__expf(-fabsf(A - C) * invt));
            Zint = fminf(Zint, mn);
            // side = softplus(2*(Zint - zint)) / 2, stable form
            float y    = 2.f * (Zint - zint);
            float side = 0.5f * (fmaxf(y, 0.f) + log1pf(__expf(-fabsf(y))));
            lv += __logf(fmaxf(side, 1e-23f));
        }
        float cond = __expf(lv - lvb[j]);
        cond = fminf(fmaxf(cond, 1e-6f), 1.f - 1e-6f);
        cond_out[(size_t)b * (Kc * Kc) + idx] = cond;
        rel += cond * Wrel[idx];
    }
    red[t] = rel;
    __syncthreads();
    #pragma unroll
    for (int s = 128; s > 0; s >>= 1) {
        if (t < s) red[t] += red[t + s];
        __syncthreads();
    }
    if (t == 0) relsum_ws[b] = red[0];
}

// ---------------------------------------------------------------------------
// Kernel 4: final_logit = boxdot + bbox + relsum + brel -> sigmoid
// ---------------------------------------------------------------------------
__global__ void __launch_bounds__(256) final_kernel(
    const float* __restrict__ boxdot, const float* __restrict__ relsum,
    const float* __restrict__ bbox,   const float* __restrict__ brel,
    float* __restrict__ task)
{
    const int b = blockIdx.x * blockDim.x + threadIdx.x;
    if (b < Bsz) task[b] = sigmoid_f(boxdot[b] + relsum[b] + bbox[0] + brel[0]);
}

// ---------------------------------------------------------------------------
extern "C" void kernel_launch(void* const* d_in, const int* in_sizes, int n_in,
                              void* d_out, int out_size, void* d_ws, size_t ws_size,
                              hipStream_t stream) {
    const float* features = (const float*)d_in[0];
    const float* Wp       = (const float*)d_in[1];
    const float* bp       = (const float*)d_in[2];
    const float* Wprob    = (const float*)d_in[3];
    const float* bprob    = (const float*)d_in[4];
    const float* Wbox     = (const float*)d_in[5];
    const float* bbox     = (const float*)d_in[6];
    const float* Wrel     = (const float*)d_in[7];
    const float* brel     = (const float*)d_in[8];

    float* out     = (float*)d_out;
    float* task    = out;                        // 4096
    float* concept = out + Bsz;                  // 4096*64
    float* cond    = concept + (size_t)Bsz * Kc; // 4096*64*64

    float* ws     = (float*)d_ws;                // ~18 MB total
    float* theta  = ws;                          // B*1024 (later z|Z)
    float* lvbox  = theta  + (size_t)Bsz * Ncols;
    float* boxdot = lvbox  + (size_t)Bsz * Kc;
    float* relsum = boxdot + Bsz;

    gemm_theta_wmma<<<dim3(Kc, Bsz / 64), 128, 0, stream>>>(features, Wp, bp, theta);
    concept_kernel<<<Bsz, 64, 0, stream>>>(theta, Wprob, bprob, Wbox,
                                           concept, lvbox, boxdot);
    pair_kernel<<<Bsz, 256, 0, stream>>>(theta, lvbox, Wrel, cond, relsum);
    final_kernel<<<(Bsz + 255) / 256, 256, 0, stream>>>(boxdot, relsum, bbox, brel, task);
}